// mMOE_39178691674290
// MI455X (gfx1250) — compile-verified
//
#include <hip/hip_runtime.h>
#include <math.h>

// ---------------------------------------------------------------------------
// MoE cross-node forward for MI455X (gfx1250, wave32, WMMA).
// Dense projections on V_WMMA_F32_16X16X4_F32; 64-row M-tiles so each weight
// (B) fetch feeds 4 WMMAs -> 4x fewer global loads per matrix op.
// ---------------------------------------------------------------------------

typedef __attribute__((ext_vector_type(2))) float v2f;
typedef __attribute__((ext_vector_type(8))) float v8f;

#define DIM   128      // model dim
#define NTOK  16384    // N*B*L tokens
#define NN    1024     // N
#define KNB   16       // neighbors per token
#define NH    8        // heads
#define DH    16       // head dim
#define ASTR  132      // padded LDS stride (132 mod 64 == 4 -> conflict-free)
#define MT    64       // rows per block (4 sub-tiles of 16)

__device__ __forceinline__ v8f wmma_k4(v2f a, v2f b, v8f c) {
  // D = A(16x4 f32) x B(4x16 f32) + C(16x16 f32)
  return __builtin_amdgcn_wmma_f32_16x16x4_f32(false, a, false, b, (short)0, c,
                                               false, false);
}

// ---------------------------------------------------------------------------
// OUT[m0+i, :] = A[m0+i, :] @ W^T (+ bias).  Block = 64 rows x 128 cols.
// 8 waves; wave w owns columns [16w,16w+16) for all 4 row sub-tiles, so each
// B float2 is reused by 4 WMMAs.
// ---------------------------------------------------------------------------
__global__ __launch_bounds__(256) void gemm_xWT(const float* __restrict__ A,
                                                const float* __restrict__ W,
                                                const float* __restrict__ bias,
                                                float* __restrict__ out) {
  __shared__ float As[MT * ASTR];
  const int m0  = blockIdx.x * MT;
  const int tid = threadIdx.x;
  {
    const int row = tid >> 2;            // 0..63
    const int c0  = (tid & 3) * 32;      // 0,32,64,96
    const float4* src = (const float4*)&A[(m0 + row) * DIM + c0];
    float* dst = &As[row * ASTR + c0];
#pragma unroll
    for (int j = 0; j < 8; ++j) {
      float4 v = src[j];
      dst[4 * j + 0] = v.x; dst[4 * j + 1] = v.y;
      dst[4 * j + 2] = v.z; dst[4 * j + 3] = v.w;
    }
  }
  __syncthreads();

  const int lane = tid & 31;
  const int n0   = (tid >> 5) * 16;      // wave's column tile
  const int nl   = lane & 15;
  const int kh   = lane >> 4;            // half-wave: K pair {0,1} vs {2,3}
  const int col  = n0 + nl;

  const float bvv = bias ? bias[col] : 0.0f;
  v8f acc[4];
#pragma unroll
  for (int s = 0; s < 4; ++s)
#pragma unroll
    for (int r = 0; r < 8; ++r) acc[s][r] = bvv;

  const float* ar   = &As[nl * ASTR];    // A-matrix: M = lane%16 (+16s)
  const float* wrow = &W[col * DIM];     // B = W^T -> B[k][col] = W[col][k]
#pragma unroll
  for (int kk = 0; kk < 32; ++kk) {
    const int kb = kk * 4 + kh * 2;
    v2f b; b.x = wrow[kb]; b.y = wrow[kb + 1];
#pragma unroll
    for (int s = 0; s < 4; ++s) {
      v2f a; a.x = ar[s * 16 * ASTR + kb]; a.y = ar[s * 16 * ASTR + kb + 1];
      acc[s] = wmma_k4(a, b, acc[s]);
    }
  }
#pragma unroll
  for (int s = 0; s < 4; ++s)
#pragma unroll
    for (int r = 0; r < 8; ++r) {
      const int m = m0 + s * 16 + r + kh * 8;   // C/D: M = r (+8 hi half)
      out[m * DIM + col] = acc[s][r];
    }
}

// ---------------------------------------------------------------------------
// Final pass: A' = BN(OPRE)*gamma+beta + x ; out += wgt_n * (A' @ W^T + b)
// Block = 64 rows = 4 consecutive n groups (wgt uniform within each 16 rows).
// ---------------------------------------------------------------------------
__global__ __launch_bounds__(256) void bn_gemm_out(
    const float* __restrict__ OP, const float* __restrict__ X,
    const float* __restrict__ W, const float* __restrict__ bias,
    const float* __restrict__ mean, const float* __restrict__ istd,
    const float* __restrict__ gamma, const float* __restrict__ beta,
    const float* __restrict__ wout, float* __restrict__ out) {
  const int nb = blockIdx.x * 4;
  const float w0 = wout[nb], w1 = wout[nb + 1];
  const float w2 = wout[nb + 2], w3 = wout[nb + 3];
  if (w0 == 0.0f && w1 == 0.0f && w2 == 0.0f && w3 == 0.0f) return;

  __shared__ float As[MT * ASTR];
  __shared__ float scale_s[DIM], shift_s[DIM];
  const int m0  = blockIdx.x * MT;
  const int tid = threadIdx.x;
  if (tid < DIM) {
    const float sc = istd[tid] * gamma[tid];
    scale_s[tid] = sc;
    shift_s[tid] = beta[tid] - mean[tid] * sc;
  }
  __syncthreads();
  {
    const int row = tid >> 2;
    const int c0  = (tid & 3) * 32;
    const int t   = m0 + row;
    const float4* so = (const float4*)&OP[t * DIM + c0];
    const float4* sx = (const float4*)&X[t * DIM + c0];
    float* dst = &As[row * ASTR + c0];
#pragma unroll
    for (int j = 0; j < 8; ++j) {
      float4 o = so[j];
      float4 xv = sx[j];
      const int d = c0 + 4 * j;
      dst[4 * j + 0] = fmaf(o.x, scale_s[d + 0], shift_s[d + 0]) + xv.x;
      dst[4 * j + 1] = fmaf(o.y, scale_s[d + 1], shift_s[d + 1]) + xv.y;
      dst[4 * j + 2] = fmaf(o.z, scale_s[d + 2], shift_s[d + 2]) + xv.z;
      dst[4 * j + 3] = fmaf(o.w, scale_s[d + 3], shift_s[d + 3]) + xv.w;
    }
  }
  __syncthreads();

  const int lane = tid & 31;
  const int n0   = (tid >> 5) * 16;
  const int nl   = lane & 15;
  const int kh   = lane >> 4;
  const int col  = n0 + nl;

  const float bvv = bias[col];
  v8f acc[4];
#pragma unroll
  for (int s = 0; s < 4; ++s)
#pragma unroll
    for (int r = 0; r < 8; ++r) acc[s][r] = bvv;

  const float* ar   = &As[nl * ASTR];
  const float* wrow = &W[col * DIM];
#pragma unroll
  for (int kk = 0; kk < 32; ++kk) {
    const int kb = kk * 4 + kh * 2;
    v2f b; b.x = wrow[kb]; b.y = wrow[kb + 1];
#pragma unroll
    for (int s = 0; s < 4; ++s) {
      v2f a; a.x = ar[s * 16 * ASTR + kb]; a.y = ar[s * 16 * ASTR + kb + 1];
      acc[s] = wmma_k4(a, b, acc[s]);
    }
  }
  const float wsub[4] = {w0, w1, w2, w3};
#pragma unroll
  for (int s = 0; s < 4; ++s)
#pragma unroll
    for (int r = 0; r < 8; ++r) {
      const int m = m0 + s * 16 + r + kh * 8;
      out[m * DIM + col] += wsub[s] * acc[s][r];
    }
}

// ---------------------------------------------------------------------------
// Attention epilogue.  k/v = tkv * (x@W^T)[gathered row] + bias (exact rewrite
// of (tkv*nf)@W^T + b).  One thread per (token, head).
// ---------------------------------------------------------------------------
__global__ __launch_bounds__(256) void attn_kernel(
    const float* __restrict__ Q, const float* __restrict__ KP,
    const float* __restrict__ VP, const float* __restrict__ tkv,
    const int* __restrict__ idx, const float* __restrict__ bk,
    const float* __restrict__ bv, float* __restrict__ out) {
  const int g  = blockIdx.x * blockDim.x + threadIdx.x;  // 0 .. NTOK*NH-1
  const int t  = g >> 3;
  const int h  = g & 7;
  const int n  = t >> 4;
  const int bl = t & 15;
  const int hb = h * DH;

  float q[DH], bkh[DH], bvh[DH];
#pragma unroll
  for (int j = 0; j < DH; ++j) {
    q[j]   = Q[t * DIM + hb + j];
    bkh[j] = bk[hb + j];
    bvh[j] = bv[hb + j];
  }

  float s[KNB], tks[KNB];
  int rows[KNB];
  float mx = -3.0e38f;
#pragma unroll
  for (int kn = 0; kn < KNB; ++kn) {
    const int r    = idx[n * KNB + kn] * 16 + bl;
    const float tk = tkv[n * KNB + kn];
    rows[kn] = r; tks[kn] = tk;
    const float* kp = &KP[r * DIM + hb];
    float acc = 0.0f;
#pragma unroll
    for (int j = 0; j < DH; ++j) acc += q[j] * (tk * kp[j] + bkh[j]);
    acc *= 0.25f;                        // 1/sqrt(dh), dh=16
    s[kn] = acc;
    mx = fmaxf(mx, acc);
  }
  float sum = 0.0f;
#pragma unroll
  for (int kn = 0; kn < KNB; ++kn) { s[kn] = expf(s[kn] - mx); sum += s[kn]; }
  const float inv = 1.0f / sum;

  float o[DH];
#pragma unroll
  for (int j = 0; j < DH; ++j) o[j] = 0.0f;
#pragma unroll
  for (int kn = 0; kn < KNB; ++kn) {
    const float p  = s[kn] * inv;
    const float tk = tks[kn];
    const float* vp = &VP[rows[kn] * DIM + hb];
#pragma unroll
    for (int j = 0; j < DH; ++j) o[j] += p * (tk * vp[j] + bvh[j]);
  }
#pragma unroll
  for (int j = 0; j < DH; ++j) out[t * DIM + hb + j] = o[j];
}

// ---------------------------------------------------------------------------
// Masked BatchNorm statistics over tokens: one block per feature d.
// ---------------------------------------------------------------------------
__global__ __launch_bounds__(256) void bn_reduce(const float* __restrict__ OP,
                                                 const float* __restrict__ mask,
                                                 const float* __restrict__ cnt,
                                                 float* __restrict__ mean,
                                                 float* __restrict__ istd) {
  const int d = blockIdx.x;
  float s = 0.0f, sq = 0.0f;
  for (int t = threadIdx.x; t < NTOK; t += 256) {
    if (mask[t >> 4] != 0.0f) {
      const float v = OP[t * DIM + d];
      s += v; sq += v * v;
    }
  }
  __shared__ float ss[256], sqq[256];
  ss[threadIdx.x] = s; sqq[threadIdx.x] = sq;
  __syncthreads();
  for (int off = 128; off > 0; off >>= 1) {
    if (threadIdx.x < off) {
      ss[threadIdx.x]  += ss[threadIdx.x + off];
      sqq[threadIdx.x] += sqq[threadIdx.x + off];
    }
    __syncthreads();
  }
  if (threadIdx.x == 0) {
    const float c   = cnt[0];
    const float mu  = ss[0] / c;
    const float var = sqq[0] / c - mu * mu;
    mean[d] = mu;
    istd[d] = 1.0f / sqrtf(var + 1e-5f);
  }
}

// ---------------------------------------------------------------------------
// Router: softmax gate, top-2, balance loss, sequential capacity cumsum,
// per-pass masks / combine weights / BN counts.  Single 1024-thread block;
// float reductions serialized on thread 0 for determinism.
// ---------------------------------------------------------------------------
__global__ __launch_bounds__(1024) void router_kernel(
    const float* __restrict__ mm, const float* __restrict__ Wr,
    const float* __restrict__ br, const int* __restrict__ modal,
    float* __restrict__ masks, float* __restrict__ wout,
    float* __restrict__ cnts, float* __restrict__ bal) {
  __shared__ float g[NN * 4];
  __shared__ unsigned char rb[NN];   // routed bitmask
  __shared__ unsigned char kp[NN];   // keep bitmask
  __shared__ float scnt[8];
  const int n = threadIdx.x;
  if (n < 8) scnt[n] = 0.0f;

  float lg[4];
  const float* xm = &mm[n * DIM];
  for (int e = 0; e < 4; ++e) {
    float acc = br[e];
    const float* w = &Wr[e * DIM];
    for (int d = 0; d < DIM; ++d) acc += xm[d] * w[d];
    lg[e] = acc;
  }
  float mx = fmaxf(fmaxf(lg[0], lg[1]), fmaxf(lg[2], lg[3]));
  float es = 0.0f;
  for (int e = 0; e < 4; ++e) { lg[e] = expf(lg[e] - mx); es += lg[e]; }
  for (int e = 0; e < 4; ++e) { lg[e] /= es; g[n * 4 + e] = lg[e]; }

  // top-2 of gate (ties -> lowest index, matches jax.lax.top_k)
  int i1 = 0;
  for (int e = 1; e < 4; ++e) if (lg[e] > lg[i1]) i1 = e;
  int i2 = -1;
  for (int e = 0; e < 4; ++e)
    if (e != i1 && (i2 < 0 || lg[e] > lg[i2])) i2 = e;
  rb[n] = (unsigned char)((1u << i1) | (1u << i2));
  __syncthreads();

  if (n == 0) {
    // deterministic serial: balance loss + capacity cumsum (N/E == 256)
    int cum[4] = {0, 0, 0, 0};
    float fs[4] = {0, 0, 0, 0}, ps[4] = {0, 0, 0, 0};
    for (int i = 0; i < NN; ++i) {
      const unsigned char r = rb[i];
      unsigned char kb = 0;
      for (int e = 0; e < 4; ++e) {
        ps[e] += g[i * 4 + e];
        if ((r >> e) & 1) {
          fs[e] += 1.0f;
          ++cum[e];
          if (cum[e] <= 256) kb |= (unsigned char)(1u << e);
        }
      }
      kp[i] = kb;
    }
    float b = 0.0f;
    for (int e = 0; e < 4; ++e) b += (fs[e] / (float)NN) * (ps[e] / (float)NN);
    bal[0] = 4.0f * b;
  }
  __syncthreads();

  const unsigned char rk = rb[n] & kp[n];
  float rp[4];
  for (int e = 0; e < 4; ++e) rp[e] = ((rk >> e) & 1) ? g[n * 4 + e] : 0.0f;
  const bool active = (rp[0] + rp[1] + rp[2] + rp[3]) > 0.0f;
  int j1 = 0;
  for (int e = 1; e < 4; ++e) if (rp[e] > rp[j1]) j1 = e;
  int j2 = -1;
  for (int e = 0; e < 4; ++e)
    if (e != j1 && (j2 < 0 || rp[e] > rp[j2])) j2 = e;

  for (int e = 0; e < 4; ++e) {
    const bool sel = active && (j1 == e || j2 == e);
    const float prob = (j1 == e ? rp[j1] : 0.0f) + (j2 == e ? rp[j2] : 0.0f);
    masks[e * NN + n] = sel ? 1.0f : 0.0f;
    wout[e * NN + n]  = sel ? prob : 0.0f;
    if (sel) atomicAdd(&scnt[e], 1.0f);    // integer-valued: exact/deterministic
  }
  for (int m2 = 0; m2 < 4; ++m2) {
    const bool sm = modal[n * 4 + m2] == 1;
    masks[(4 + m2) * NN + n] = sm ? 1.0f : 0.0f;
    wout[(4 + m2) * NN + n]  = sm ? 0.25f : 0.0f;  // fold 1/Mn into weight
    if (sm) atomicAdd(&scnt[4 + m2], 1.0f);
  }
  __syncthreads();
  if (n < 8) cnts[n] = fmaxf(scnt[n] * 16.0f, 1.0f);  // *BL, clamp like ref
}

__global__ void zero_kernel(float* p, int nelem) {
  const int i = blockIdx.x * blockDim.x + threadIdx.x;
  if (i < nelem) p[i] = 0.0f;
}

// ---------------------------------------------------------------------------
extern "C" void kernel_launch(void* const* d_in, const int* in_sizes, int n_in,
                              void* d_out, int out_size, void* d_ws,
                              size_t ws_size, hipStream_t stream) {
  (void)in_sizes; (void)n_in; (void)out_size; (void)ws_size;

  const float* x     = (const float*)d_in[0];
  const float* mm    = (const float*)d_in[1];
  const float* tkv   = (const float*)d_in[2];
  const int*   idx   = (const int*)d_in[3];
  const int*   modal = (const int*)d_in[4];
  const float* Wr    = (const float*)d_in[5];
  const float* br    = (const float*)d_in[6];
  const float* eW[5]; for (int j = 0; j < 5; ++j) eW[j] = (const float*)d_in[7 + j];
  const float* mW[5]; for (int j = 0; j < 5; ++j) mW[j] = (const float*)d_in[12 + j];
  const float* eb[6]; for (int j = 0; j < 6; ++j) eb[j] = (const float*)d_in[17 + j]; // bq,bk,bv,bo,Beta,b1
  const float* mb[6]; for (int j = 0; j < 6; ++j) mb[j] = (const float*)d_in[23 + j]; // bq,bk,bv,bo,Beta,b2
  const float* eGam  = (const float*)d_in[29];
  const float* mGam  = (const float*)d_in[30];

  float* out = (float*)d_out;                 // [16384,128] + balance_loss
  float* ws  = (float*)d_ws;
  float* Q     = ws;
  float* KP    = Q   + (size_t)NTOK * DIM;
  float* VP    = KP  + (size_t)NTOK * DIM;
  float* AT    = VP  + (size_t)NTOK * DIM;
  float* OP    = AT  + (size_t)NTOK * DIM;
  float* MASKS = OP  + (size_t)NTOK * DIM;    // [8][1024]
  float* WOUT  = MASKS + 8 * NN;              // [8][1024]
  float* CNTS  = WOUT + 8 * NN;               // [8]
  float* MEAN  = CNTS + 8;                    // [128]
  float* ISTD  = MEAN + DIM;                  // [128]

  const int total = NTOK * DIM + 1;
  zero_kernel<<<(total + 255) / 256, 256, 0, stream>>>(out, total);
  router_kernel<<<1, 1024, 0, stream>>>(mm, Wr, br, modal, MASKS, WOUT, CNTS,
                                        out + (size_t)NTOK * DIM);

  for (int p = 0; p < 8; ++p) {
    const int i = (p < 4) ? p : (p - 4);
    const float* Wq = ((p < 4) ? eW[0] : mW[0]) + (size_t)i * DIM * DIM;
    const float* Wk = ((p < 4) ? eW[1] : mW[1]) + (size_t)i * DIM * DIM;
    const float* Wv = ((p < 4) ? eW[2] : mW[2]) + (size_t)i * DIM * DIM;
    const float* Wo = ((p < 4) ? eW[3] : mW[3]) + (size_t)i * DIM * DIM;
    const float* Wl = ((p < 4) ? eW[4] : mW[4]) + (size_t)i * DIM * DIM;
    const float* bq = ((p < 4) ? eb[0] : mb[0]) + (size_t)i * DIM;
    const float* bk = ((p < 4) ? eb[1] : mb[1]) + (size_t)i * DIM;
    const float* bv = ((p < 4) ? eb[2] : mb[2]) + (size_t)i * DIM;
    const float* bo = ((p < 4) ? eb[3] : mb[3]) + (size_t)i * DIM;
    const float* bt = ((p < 4) ? eb[4] : mb[4]) + (size_t)i * DIM;  // beta
    const float* bl = ((p < 4) ? eb[5] : mb[5]) + (size_t)i * DIM;
    const float* gm = ((p < 4) ? eGam : mGam)  + (size_t)i * DIM;

    gemm_xWT<<<NTOK / MT, 256, 0, stream>>>(x, Wq, bq, Q);
    gemm_xWT<<<NTOK / MT, 256, 0, stream>>>(x, Wk, nullptr, KP);
    gemm_xWT<<<NTOK / MT, 256, 0, stream>>>(x, Wv, nullptr, VP);
    attn_kernel<<<(NTOK * NH) / 256, 256, 0, stream>>>(Q, KP, VP, tkv, idx, bk,
                                                       bv, AT);
    gemm_xWT<<<NTOK / MT, 256, 0, stream>>>(AT, Wo, bo, OP);
    bn_reduce<<<DIM, 256, 0, stream>>>(OP, MASKS + (size_t)p * NN,
                                       CNTS + p, MEAN, ISTD);
    bn_gemm_out<<<NTOK / MT, 256, 0, stream>>>(OP, x, Wl, bl, MEAN, ISTD, gm,
                                               bt, WOUT + (size_t)p * NN, out);
  }
}